// OptimizedQuantumShadowNoise_42374147342479
// MI455X (gfx1250) — compile-verified
//
#include <hip/hip_runtime.h>
#include <hip/hip_bf16.h>

#define NQ   10
#define DIM  1024
#define NL   2
#define NB   3
#define ZD   100
#define TPB  256
#define APT  (DIM / TPB)      // amplitudes per thread = 4
#define PPT  (DIM / 2 / TPB)  // butterfly pairs per thread per gate = 2

typedef __attribute__((ext_vector_type(2))) float v2f;
typedef __attribute__((ext_vector_type(8))) float v8f;

// ---------------------------------------------------------------------------
// Kernel 1: per-sample 10-qubit state-vector simulation, state in LDS.
// One block = one sample. Writes the 3 Pauli expectations (K-padded to 4)
// into workspace for the WMMA projection kernel.
// ---------------------------------------------------------------------------
__global__ __launch_bounds__(TPB)
void qsim_kernel(const float* __restrict__ z,     // [2, B]
                 const float* __restrict__ W,     // [NL, 2*NQ-1]
                 const int*   __restrict__ pidx,  // [NB, NQ]  0=Z 1=X 2=Y 3=I
                 float* __restrict__ shadow,      // [B, 4] (k=3 zero pad)
                 int B)
{
    __shared__ float sre[DIM];
    __shared__ float sim_[DIM];
    __shared__ float red[(TPB / 32) * NB];

    const int b   = blockIdx.x;
    const int tid = threadIdx.x;

    // ---- per-sample angles ----
    const float a  = asinf(z[b]);
    const float bb = asinf(z[B + b]);
    float sa, ca;
    __sincosf(0.5f * a, &sa, &ca);

    // ---- analytic product-state init: RY(a)^n |0..0> then RZ(b) diagonal ----
    #pragma unroll
    for (int j = 0; j < APT; ++j) {
        const int idx = tid + TPB * j;
        float m = 1.0f;
        #pragma unroll
        for (int w = 0; w < NQ; ++w)
            m *= ((idx >> (NQ - 1 - w)) & 1) ? sa : ca;
        const int p = __popc(idx);
        const float phi = 0.5f * bb * (float)(2 * p - NQ);
        float sphi, cphi;
        __sincosf(phi, &sphi, &cphi);
        sre[idx]  = m * cphi;
        sim_[idx] = m * sphi;
    }

    // ---- layers: 10 RY butterflies + diagonal ZZ phase ----
    for (int layer = 0; layer < NL; ++layer) {
        const float* Wl = W + layer * (2 * NQ - 1);

        for (int w = 0; w < NQ; ++w) {
            __syncthreads();
            float sth, cth;
            __sincosf(0.5f * Wl[w], &sth, &cth);
            const int k = NQ - 1 - w;          // bit position of wire w
            #pragma unroll
            for (int q = 0; q < PPT; ++q) {
                const int pp = tid + q * TPB;  // pair id 0..511
                const int i0 = ((pp >> k) << (k + 1)) | (pp & ((1 << k) - 1));
                const int i1 = i0 | (1 << k);
                const float x0r = sre[i0], x0i = sim_[i0];
                const float x1r = sre[i1], x1i = sim_[i1];
                sre[i0]  = cth * x0r - sth * x1r;
                sim_[i0] = cth * x0i - sth * x1i;
                sre[i1]  = sth * x0r + cth * x1r;
                sim_[i1] = sth * x0i + cth * x1i;
            }
        }
        __syncthreads();

        float wz[NQ - 1];
        #pragma unroll
        for (int i = 0; i < NQ - 1; ++i) wz[i] = Wl[NQ + i];

        #pragma unroll
        for (int j = 0; j < APT; ++j) {
            const int idx = tid + TPB * j;
            float ang = 0.0f;
            #pragma unroll
            for (int i = 0; i < NQ - 1; ++i) {
                const int x = (idx >> (NQ - 1 - i)) ^ (idx >> (NQ - 2 - i));
                ang += (x & 1) ? -wz[i] : wz[i];
            }
            float sA, cA;
            __sincosf(0.5f * ang, &sA, &cA);     // psi *= exp(-i*ang/2)
            const float re = sre[idx], im = sim_[idx];
            sre[idx]  = re * cA + im * sA;
            sim_[idx] = im * cA - re * sA;
        }
    }
    __syncthreads();

    // ---- Pauli-string expectations via flip/sign masks ----
    int flipM[NB], signM[NB], ny[NB];
    #pragma unroll
    for (int kk = 0; kk < NB; ++kk) {
        int fm = 0, sm = 0, cy = 0;
        for (int w = 0; w < NQ; ++w) {
            const int p   = pidx[kk * NQ + w];
            const int bit = 1 << (NQ - 1 - w);
            if (p == 1 || p == 2) fm |= bit;     // X or Y flips
            if (p == 0 || p == 2) sm |= bit;     // Z or Y sign
            if (p == 2) cy++;                    // each Y contributes (-i)
        }
        flipM[kk] = fm; signM[kk] = sm; ny[kk] = cy & 3;
    }

    float acc[NB] = {0.0f, 0.0f, 0.0f};
    #pragma unroll
    for (int j = 0; j < APT; ++j) {
        const int x = tid + TPB * j;
        const float ar = sre[x], ai = sim_[x];
        #pragma unroll
        for (int kk = 0; kk < NB; ++kk) {
            const int y = x ^ flipM[kk];
            const float br = sre[y], bi = sim_[y];
            const float dot = ar * br + ai * bi;   // Re(conj(a)*b)
            const float crs = ar * bi - ai * br;   // Im(conj(a)*b)
            const int m4 = ny[kk];
            float t = (m4 == 0) ?  dot :
                      (m4 == 1) ?  crs :
                      (m4 == 2) ? -dot : -crs;     // times (-i)^nY, real part
            const float sgn = (__popc(x & signM[kk]) & 1) ? -1.0f : 1.0f;
            acc[kk] += sgn * t;
        }
    }

    // ---- wave32 reduce + cross-wave reduce ----
    #pragma unroll
    for (int kk = 0; kk < NB; ++kk) {
        float v = acc[kk];
        for (int off = 16; off > 0; off >>= 1)
            v += __shfl_xor(v, off, 32);
        if ((tid & 31) == 0) red[(tid >> 5) * NB + kk] = v;
    }
    __syncthreads();
    if (tid == 0) {
        #pragma unroll
        for (int kk = 0; kk < NB; ++kk) {
            float s = 0.0f;
            for (int wv = 0; wv < TPB / 32; ++wv) s += red[wv * NB + kk];
            shadow[b * 4 + kk] = s;
        }
        shadow[b * 4 + 3] = 0.0f;                  // K pad for WMMA
    }
}

// ---------------------------------------------------------------------------
// Kernel 2: out[B,100] = shadow[B,3] @ coeffs[3,100] via V_WMMA_F32_16X16X4_F32
// (K padded to 4, N padded to 112). One wave per 16x16 tile.
// A layout (f32 16x4): lanes 0-15 -> M=lane, K={0,1}; lanes 16-31 -> K={2,3}.
// B layout (f32 4x16): lanes 0-15 -> N=lane, K={0,1}; lanes 16-31 -> K={2,3}.
// D layout: VGPR r -> row r (lanes 0-15) / row r+8 (lanes 16-31), N=lane&15.
// ---------------------------------------------------------------------------
__global__ __launch_bounds__(32)
void shadow_gemm_wmma(const float* __restrict__ shadow,  // [B,4]
                      const float* __restrict__ coeffs,  // [NB, ZD]
                      float* __restrict__ out,           // [B, ZD]
                      int B)
{
    const int lane = threadIdx.x;
    const int m0 = blockIdx.x * 16;
    const int n0 = blockIdx.y * 16;
    const int r  = lane & 15;
    const int kb = (lane >> 4) << 1;          // 0 or 2

    v2f A;
    const int row = m0 + r;
    A.x = shadow[row * 4 + kb];
    A.y = shadow[row * 4 + kb + 1];

    v2f Bm;
    const int col = n0 + r;
    Bm.x = (col < ZD && (kb    ) < NB) ? coeffs[(kb    ) * ZD + col] : 0.0f;
    Bm.y = (col < ZD && (kb + 1) < NB) ? coeffs[(kb + 1) * ZD + col] : 0.0f;

    v8f C = {};
    v8f D = __builtin_amdgcn_wmma_f32_16x16x4_f32(
        /*neg_a=*/false, A, /*neg_b=*/false, Bm,
        /*c_mod=*/(short)0, C, /*reuse_a=*/false, /*reuse_b=*/false);

    const int mAdd = (lane >> 4) << 3;        // +8 for upper half-wave
    if (col < ZD) {
        #pragma unroll
        for (int rr = 0; rr < 8; ++rr) {
            const int m = m0 + rr + mAdd;
            out[m * ZD + col] = D[rr];
        }
    }
}

// ---------------------------------------------------------------------------
extern "C" void kernel_launch(void* const* d_in, const int* in_sizes, int n_in,
                              void* d_out, int out_size, void* d_ws, size_t ws_size,
                              hipStream_t stream) {
    const float* z      = (const float*)d_in[0];   // [2, B]
    const float* W      = (const float*)d_in[1];   // [NL, 19]
    const float* coeffs = (const float*)d_in[2];   // [3, 100]
    const int*   pidx   = (const int*)  d_in[3];   // [3, 10]
    float*       out    = (float*)d_out;           // [B, 100]
    const int    B      = in_sizes[0] / 2;         // 4096

    float* shadow = (float*)d_ws;                  // [B, 4], 64 KB

    qsim_kernel<<<B, TPB, 0, stream>>>(z, W, pidx, shadow, B);

    dim3 grid(B / 16, (ZD + 15) / 16);             // 256 x 7 tiles
    shadow_gemm_wmma<<<grid, 32, 0, stream>>>(shadow, coeffs, out, B);
}